// Model_79748952752243
// MI455X (gfx1250) — compile-verified
//
#include <hip/hip_runtime.h>
#include <math.h>

// ----------------------------------------------------------------------------
// Gibbs sweep over a (B,32) Gaussian state, MI455X / gfx1250.
//
// Algebra: the sequential coordinate sweep is Gauss-Seidel:
//   (D+L) x_new = D*mu - U*x_old + D*s (*) n ,  s_i = sqrt(1/P_ii)
// With M = (D+L)^-1 (32x32 lower triangular, fixed per call):
//   x_new = x_old @ A^T + n @ G^T + c ,  A=-M*U, G=M*D*diag(s), c=M*D*mu
// => one (Bx64)@(64x32) fp32 GEMM. Memory bound: 384 MB @ 23.3 TB/s ~ 16.5us;
// 4.3 GFLOP is noise, so fp32 WMMA (16x16x4) keeps full reference precision.
// ----------------------------------------------------------------------------

typedef float v2f __attribute__((ext_vector_type(2)));
typedef float v8f __attribute__((ext_vector_type(8)));

#define DIM 32

// ---------------- prep: build W = [A | G] (32x64 row-major) and c (32) -------
__global__ void gibbs_prep(const float* __restrict__ P, const float* __restrict__ mu,
                           float* __restrict__ W, float* __restrict__ cvec)
{
    __shared__ float Ps[DIM][DIM];
    __shared__ float M[DIM][DIM];
    const int t = threadIdx.x;                 // 0..31, one wave
    for (int j = 0; j < DIM; ++j) Ps[t][j] = P[t * DIM + j];
    __syncthreads();

    // Thread t solves (D+L) m = e_t  (column t of M): independent per thread.
    for (int i = 0; i < DIM; ++i) {
        if (i < t) { M[i][t] = 0.0f; continue; }
        float v = (i == t) ? 1.0f : 0.0f;
        for (int j = t; j < i; ++j) v -= Ps[i][j] * M[j][t];
        M[i][t] = v / Ps[i][i];
    }
    __syncthreads();

    // Thread t emits row t of W and c[t].
    //   A[t][j] = -sum_{l < j, l <= t} M[t][l] * P[l][j]
    //   G[t][j] =  M[t][j] * sqrt(P[j][j])          (j <= t)
    //   c[t]    =  sum_j M[t][j] * P[j][j] * mu[j]
    float ci = 0.0f;
    for (int j = 0; j < DIM; ++j) {
        float a = 0.0f;
        const int lmax = (j - 1 < t) ? (j - 1) : t;
        for (int l = 0; l <= lmax; ++l) a -= M[t][l] * Ps[l][j];
        W[t * 2 * DIM + j] = a;
        W[t * 2 * DIM + DIM + j] = (j <= t) ? M[t][j] * sqrtf(Ps[j][j]) : 0.0f;
        ci += M[t][j] * Ps[j][j] * mu[j];
    }
    cvec[t] = ci;
}

// ---------------- main GEMM: out = [x | n] @ W^T + c -------------------------
// Per wave: 16 batch rows (M), 2 N-tiles of 16 coords, K=64 in 16 steps of 4.
// W^T B-fragments are loaded once and stay resident in VGPRs across the
// grid-stride loop. A-fragment (16x4 f32) layout per ISA: lanes 0-15 hold
// K={k0,k0+1}, lanes 16-31 hold K={k0+2,k0+3} for rows M=lane%16.
__global__ void __launch_bounds__(256)
gibbs_sweep(const float* __restrict__ x, const float* __restrict__ noise,
            const float* __restrict__ W, const float* __restrict__ cvec,
            float* __restrict__ out, int nRowBlocks)
{
    const int lane = threadIdx.x & 31;
    const int half = lane >> 4;          // selects K pair within a fragment
    const int m    = lane & 15;          // A: row-in-tile, B/D: N-in-tile
    const int wavesPerBlock = blockDim.x >> 5;
    const int waveId = blockIdx.x * wavesPerBlock + (threadIdx.x >> 5);
    const int nWaves = gridDim.x * wavesPerBlock;

    // Resident B-fragments of W^T: wf0 -> coords 0..15, wf1 -> coords 16..31.
    v2f wf0[16], wf1[16];
#pragma unroll
    for (int t = 0; t < 16; ++t) {
        const int col = 4 * t + 2 * half;
        wf0[t] = *(const v2f*)(W + (m     ) * 64 + col);
        wf1[t] = *(const v2f*)(W + (m + 16) * 64 + col);
    }
    const float c0 = cvec[m];
    const float c1 = cvec[m + 16];

    for (int rb = waveId; rb < nRowBlocks; rb += nWaves) {
        const long rowBase = (long)rb * 16;
        const float* xr = x     + (rowBase + m) * DIM;
        const float* nr = noise + (rowBase + m) * DIM;
        v8f acc0 = {0.f, 0.f, 0.f, 0.f, 0.f, 0.f, 0.f, 0.f};
        v8f acc1 = acc0;
#pragma unroll
        for (int t = 0; t < 8; ++t) {            // K = 0..31 : x coefficients
            v2f a = *(const v2f*)(xr + 4 * t + 2 * half);
            acc0 = __builtin_amdgcn_wmma_f32_16x16x4_f32(false, a, false, wf0[t],
                                                         (short)0, acc0, false, false);
            acc1 = __builtin_amdgcn_wmma_f32_16x16x4_f32(false, a, false, wf1[t],
                                                         (short)0, acc1, false, false);
        }
#pragma unroll
        for (int t = 0; t < 8; ++t) {            // K = 32..63 : noise coeffs
            v2f a = *(const v2f*)(nr + 4 * t + 2 * half);
            acc0 = __builtin_amdgcn_wmma_f32_16x16x4_f32(false, a, false, wf0[t + 8],
                                                         (short)0, acc0, false, false);
            acc1 = __builtin_amdgcn_wmma_f32_16x16x4_f32(false, a, false, wf1[t + 8],
                                                         (short)0, acc1, false, false);
        }
        // D tile: lanes 0-15 -> rows v, lanes 16-31 -> rows 8+v; N = m.
        float* orow = out + rowBase * DIM;
#pragma unroll
        for (int v = 0; v < 8; ++v) {
            const int r = v + 8 * half;
            orow[r * DIM + m]      = acc0[v] + c0;
            orow[r * DIM + m + 16] = acc1[v] + c1;
        }
    }
}

extern "C" void kernel_launch(void* const* d_in, const int* in_sizes, int n_in,
                              void* d_out, int out_size, void* d_ws, size_t ws_size,
                              hipStream_t stream) {
    const float* x     = (const float*)d_in[0];
    const float* noise = (const float*)d_in[1];
    const float* P     = (const float*)d_in[2];
    const float* mu    = (const float*)d_in[3];
    float* out  = (float*)d_out;
    float* W    = (float*)d_ws;              // 32*64 floats
    float* cvec = W + DIM * 2 * DIM;         // 32 floats

    const int rows = in_sizes[0] / DIM;      // B = 1,048,576
    const int nRowBlocks = rows / 16;        // 65,536 16-row tiles

    gibbs_prep<<<1, 32, 0, stream>>>(P, mu, W, cvec);
    gibbs_sweep<<<2048, 256, 0, stream>>>(x, noise, W, cvec, out, nRowBlocks);
}